// PosRegressor_43482248904949
// MI455X (gfx1250) — compile-verified
//
#include <hip/hip_runtime.h>

// Bilinear sub-pixel shift with mirror boundaries, 256 x 512 x 512 f32.
// Memory-bound (≈512 MB traffic → ~22 µs roofline @ 23.3 TB/s).
// Uses CDNA5 async global->LDS copies (ASYNCcnt) to stage mirrored input
// rows once per row-block, then computes 4-wide vectorized outputs.

#define IMG_H 512
#define IMG_W 512
#define ROWS_PER_BLOCK 8
#define NTHREADS 256

__device__ __forceinline__ int mirror_idx(int idx) {
  // p = 2*(n-1) = 1022 for n = 512
  int i = abs(idx) % (2 * (IMG_H - 1));
  return (i >= IMG_H) ? (2 * (IMG_H - 1) - i) : i;
}

// Flat (generic) LDS address -> 32-bit LDS offset: low 32 bits of the flat
// address are the LDS byte offset.
__device__ __forceinline__ unsigned lds_off32(const void* p) {
  return (unsigned)(unsigned long long)p;
}

__global__ __launch_bounds__(NTHREADS) void shift_bilinear_kernel(
    const float* __restrict__ images, const float* __restrict__ dxdy,
    float* __restrict__ out) {
  __shared__ float tile[2 * ROWS_PER_BLOCK][IMG_W];  // 32 KB

  const int img = blockIdx.y;
  const int r0 = blockIdx.x * ROWS_PER_BLOCK;
  const float dy = dxdy[2 * img + 0];
  const float dx = dxdy[2 * img + 1];
  const float* __restrict__ src = images + (size_t)img * IMG_H * IMG_W;
  float* __restrict__ dst = out + (size_t)img * IMG_H * IMG_W;

  const int t = threadIdx.x;
  const int lane_row = t >> 7;     // 0 -> y0m row, 1 -> y1m row
  const int col4 = (t & 127) * 4;  // starting column (float index), 16B aligned

  // ---- Stage 2*ROWS_PER_BLOCK mirrored input rows into LDS (async path) ----
#pragma unroll
  for (int p = 0; p < ROWS_PER_BLOCK; ++p) {
    const int r = r0 + p;
    const float cy = (float)r - dy;
    const int y0 = (int)floorf(cy);
    const int ysrc = mirror_idx(y0 + lane_row);
    const unsigned long long ga =
        (unsigned long long)(const void*)(src + (size_t)ysrc * IMG_W + col4);
    const unsigned la = lds_off32(&tile[2 * p + lane_row][col4]);
    asm volatile("global_load_async_to_lds_b128 %0, %1, off"
                 :: "v"(la), "v"(ga)
                 : "memory");
  }
#if __has_builtin(__builtin_amdgcn_s_wait_asynccnt)
  __builtin_amdgcn_s_wait_asynccnt(0);
#else
  asm volatile("s_wait_asynccnt 0x0" ::: "memory");
#endif
  __syncthreads();

  // ---- Column-only weights/indices (constant across rows of this image) ----
  float fx[4];
  int x0m[4], x1m[4];
#pragma unroll
  for (int k = 0; k < 4; ++k) {
    const int x = col4 + k;
    const float cx = (float)x - dx;
    const float x0f = floorf(cx);
    fx[k] = cx - x0f;
    const int x0 = (int)x0f;
    x0m[k] = mirror_idx(x0);
    x1m[k] = mirror_idx(x0 + 1);
  }

  // ---- Compute: 4 passes x 2 rows, 4 columns per thread, b128 stores ----
#pragma unroll
  for (int pass = 0; pass < ROWS_PER_BLOCK / 2; ++pass) {
    const int rl = pass * 2 + lane_row;  // local row 0..7
    const int r = r0 + rl;
    const float cy = (float)r - dy;
    const float fy = cy - floorf(cy);
    const float wy0 = 1.0f - fy;
    float res[4];
#pragma unroll
    for (int k = 0; k < 4; ++k) {
      const float v00 = tile[2 * rl + 0][x0m[k]];
      const float v01 = tile[2 * rl + 0][x1m[k]];
      const float v10 = tile[2 * rl + 1][x0m[k]];
      const float v11 = tile[2 * rl + 1][x1m[k]];
      const float wx0 = 1.0f - fx[k];
      const float top = wx0 * v00 + fx[k] * v01;
      const float bot = wx0 * v10 + fx[k] * v11;
      res[k] = wy0 * top + fy * bot;
    }
    *reinterpret_cast<float4*>(dst + (size_t)r * IMG_W + col4) =
        make_float4(res[0], res[1], res[2], res[3]);
  }
}

extern "C" void kernel_launch(void* const* d_in, const int* in_sizes, int n_in,
                              void* d_out, int out_size, void* d_ws, size_t ws_size,
                              hipStream_t stream) {
  const float* images = (const float*)d_in[0];
  const float* dxdy = (const float*)d_in[1];
  float* out = (float*)d_out;
  const int n_img = (n_in > 1) ? (in_sizes[1] / 2) : 256;  // 256 images
  dim3 grid(IMG_H / ROWS_PER_BLOCK, n_img);
  shift_bilinear_kernel<<<grid, NTHREADS, 0, stream>>>(images, dxdy, out);
}